// ResidualBlock_48644799595039
// MI455X (gfx1250) — compile-verified
//
#include <hip/hip_runtime.h>

typedef float v2f __attribute__((ext_vector_type(2)));
typedef float v8f __attribute__((ext_vector_type(8)));

__device__ __forceinline__ float leaky(float v) {
    // leaky_relu(v, 0.1): v>=0 -> v ; v<0 -> 0.1v  == max(v, 0.1v)
    return fmaxf(v, 0.1f * v);
}

// One wave computes a 16-row x 64-col tile of  leaky(X @ W^T + bias).
// X: [M, Kd] row-major, W: [Nout, Kd] row-major (so B-frag = rows of W).
// res[t][r] = value for row (m0 + r + 8*(lane/16)), col (n0 + 16*t + lane%16).
__device__ __forceinline__ void gemm_tile_16x64(const float* __restrict__ X,
                                                const float* __restrict__ W,
                                                const float* __restrict__ bias,
                                                int Kd, int m0, int n0,
                                                float res[4][8]) {
    const int lane = threadIdx.x & 31;
    const int half = lane >> 4;   // 0: K pair {0,1}, 1: K pair {2,3}
    const int lm   = lane & 15;

    v8f acc0 = {}, acc1 = {}, acc2 = {}, acc3 = {};

    const float* ap = X + (size_t)(m0 + lm) * Kd + 2 * half;
    const float* bp = W + (size_t)(n0 + lm) * Kd + 2 * half;
    const size_t wrow16 = (size_t)16 * Kd;

    for (int k0 = 0; k0 < Kd; k0 += 4) {
        v2f a  = *(const v2f*)(ap + k0);
        v2f b0 = *(const v2f*)(bp + k0);
        v2f b1 = *(const v2f*)(bp + wrow16 + k0);
        v2f b2 = *(const v2f*)(bp + 2 * wrow16 + k0);
        v2f b3 = *(const v2f*)(bp + 3 * wrow16 + k0);
        acc0 = __builtin_amdgcn_wmma_f32_16x16x4_f32(false, a, false, b0, (short)0, acc0, false, false);
        acc1 = __builtin_amdgcn_wmma_f32_16x16x4_f32(false, a, false, b1, (short)0, acc1, false, false);
        acc2 = __builtin_amdgcn_wmma_f32_16x16x4_f32(false, a, false, b2, (short)0, acc2, false, false);
        acc3 = __builtin_amdgcn_wmma_f32_16x16x4_f32(false, a, false, b3, (short)0, acc3, false, false);
    }

    float bb0 = bias[n0 + 0 * 16 + lm];
    float bb1 = bias[n0 + 1 * 16 + lm];
    float bb2 = bias[n0 + 2 * 16 + lm];
    float bb3 = bias[n0 + 3 * 16 + lm];
#pragma unroll
    for (int r = 0; r < 8; ++r) {
        res[0][r] = leaky(acc0[r] + bb0);
        res[1][r] = leaky(acc1[r] + bb1);
        res[2][r] = leaky(acc2[r] + bb2);
        res[3][r] = leaky(acc3[r] + bb3);
    }
}

// Kernel 1: x1 = leaky(input @ W_init^T + b_init)   [M,256] -> [M,64]
__global__ void k_init_mlp(const float* __restrict__ X,
                           const float* __restrict__ W,
                           const float* __restrict__ b,
                           float* __restrict__ Y, int M) {
    const int wave = (int)((blockIdx.x * blockDim.x + threadIdx.x) >> 5);
    const int numTiles = M >> 4;
    if (wave >= numTiles) return;   // wave-uniform: EXEC stays all-ones
    const int m0 = wave << 4;

    float res[4][8];
    gemm_tile_16x64(X, W, b, 256, m0, 0, res);

    const int lane = threadIdx.x & 31;
    const int half = lane >> 4;
    const int lm   = lane & 15;
#pragma unroll
    for (int t = 0; t < 4; ++t)
#pragma unroll
        for (int r = 0; r < 8; ++r)
            Y[(size_t)(m0 + r + 8 * half) * 64 + t * 16 + lm] = res[t][r];
}

// Kernels 2/3: LFA stage. One thread per (point p, channel c), c in [0,C).
//  out[p][c]     = mean_k leaky(geom[p][k] . Wl[c] + bl[c])
//  out[p][C + c] = mean_k fin[batch_base + idx[p][k]][c]
__global__ void k_lfa(const float* __restrict__ fin,   // [M, C]
                      const float* __restrict__ geom,  // [M, 16, 4]
                      const int*   __restrict__ idx,   // [M, 16]
                      const float* __restrict__ Wl,    // [C, 4]
                      const float* __restrict__ bl,    // [C]
                      float* __restrict__ out,         // [M, 2C]
                      int M, int Nper, int C) {
    const int g = blockIdx.x * blockDim.x + threadIdx.x;
    const int p = g / C;
    const int c = g - p * C;
    if (p >= M) return;
    const int batch_base = (p / Nper) * Nper;

    const float4 w  = *(const float4*)(Wl + (size_t)c * 4);
    const float  bb = bl[c];

    const float4* gp = (const float4*)(geom + (size_t)p * 64);
    const int*    ip = idx + (size_t)p * 16;

    float ts = 0.0f, fs = 0.0f;
#pragma unroll
    for (int k = 0; k < 16; ++k) {
        const float4 gg = gp[k];
        float v = fmaf(gg.x, w.x, fmaf(gg.y, w.y, fmaf(gg.z, w.z, fmaf(gg.w, w.w, bb))));
        ts += leaky(v);
        const int row = batch_base + ip[k];
        fs += fin[(size_t)row * C + c];
    }
    out[(size_t)p * (2 * C) + c]     = ts * 0.0625f;
    out[(size_t)p * (2 * C) + C + c] = fs * 0.0625f;
}

// Kernel 4: out = leaky(x3 @ W_fin^T + b_fin) + leaky(input @ W_id^T + b_id)
// One wave per (row-tile, 64-col group); Nout = 256 -> 4 col groups.
__global__ void k_final(const float* __restrict__ X3,
                        const float* __restrict__ Wf,
                        const float* __restrict__ bf,
                        const float* __restrict__ X0,
                        const float* __restrict__ Wi,
                        const float* __restrict__ bi,
                        float* __restrict__ OUT, int M) {
    const int wave = (int)((blockIdx.x * blockDim.x + threadIdx.x) >> 5);
    const int numRowTiles = M >> 4;
    if (wave >= numRowTiles * 4) return;  // wave-uniform
    const int rt = wave >> 2;
    const int cg = wave & 3;
    const int m0 = rt << 4;
    const int n0 = cg << 6;

    float r1[4][8], r2[4][8];
    gemm_tile_16x64(X3, Wf, bf, 256, m0, n0, r1);
    gemm_tile_16x64(X0, Wi, bi, 256, m0, n0, r2);

    const int lane = threadIdx.x & 31;
    const int half = lane >> 4;
    const int lm   = lane & 15;
#pragma unroll
    for (int t = 0; t < 4; ++t)
#pragma unroll
        for (int r = 0; r < 8; ++r)
            OUT[(size_t)(m0 + r + 8 * half) * 256 + n0 + t * 16 + lm] =
                r1[t][r] + r2[t][r];
}

extern "C" void kernel_launch(void* const* d_in, const int* in_sizes, int n_in,
                              void* d_out, int out_size, void* d_ws, size_t ws_size,
                              hipStream_t stream) {
    const float* input  = (const float*)d_in[0];   // [B,N,256]
    const float* geom   = (const float*)d_in[1];   // [B,N,16,4]
    const int*   idx    = (const int*)  d_in[2];   // [B,N,16]
    const float* W_init = (const float*)d_in[3];   // [64,256]
    const float* b_init = (const float*)d_in[4];
    const float* W_l1   = (const float*)d_in[5];   // [64,4]
    const float* b_l1   = (const float*)d_in[6];
    const float* W_l2   = (const float*)d_in[7];   // [128,4]
    const float* b_l2   = (const float*)d_in[8];
    const float* W_fin  = (const float*)d_in[9];   // [256,256]
    const float* b_fin  = (const float*)d_in[10];
    const float* W_id   = (const float*)d_in[11];  // [256,256]
    const float* b_id   = (const float*)d_in[12];
    float* out = (float*)d_out;

    const int M    = in_sizes[0] / 256;  // B*N = 40000
    const int B    = 2;                  // reference constant
    const int Nper = M / B;

    // Workspace layout (fp32): x1 [M,64] | x2 [M,128] | x3 [M,256]  (~72 MB)
    float* x1 = (float*)d_ws;
    float* x2 = x1 + (size_t)M * 64;
    float* x3 = x2 + (size_t)M * 128;

    const int BLK = 256;  // 8 waves per block

    // 1) x1 = leaky(input @ W_init^T + b_init)
    {
        int waves = M >> 4;                       // 2500 row tiles
        int blocks = (waves + 7) / 8;
        k_init_mlp<<<blocks, BLK, 0, stream>>>(input, W_init, b_init, x1, M);
    }
    // 2) x2 = [mean leaky(geom@W_l1^T+b), mean gather(x1)]  -> [M,128]
    {
        int threads = M * 64;
        k_lfa<<<(threads + BLK - 1) / BLK, BLK, 0, stream>>>(
            x1, geom, idx, W_l1, b_l1, x2, M, Nper, 64);
    }
    // 3) x3 = [mean leaky(geom@W_l2^T+b), mean gather(x2)]  -> [M,256]
    {
        int threads = M * 128;
        k_lfa<<<(threads + BLK - 1) / BLK, BLK, 0, stream>>>(
            x2, geom, idx, W_l2, b_l2, x3, M, Nper, 128);
    }
    // 4) out = leaky(x3 @ W_fin^T + b_fin) + leaky(input @ W_id^T + b_id)
    {
        int waves = (M >> 4) * 4;                 // 10000
        int blocks = (waves + 7) / 8;
        k_final<<<blocks, BLK, 0, stream>>>(x3, W_fin, b_fin, input, W_id, b_id, out, M);
    }
    (void)n_in; (void)out_size; (void)ws_size;
}